// MMSEPICDetector_39616778338617
// MI455X (gfx1250) — compile-verified
//
#include <hip/hip_runtime.h>
#include <math.h>

// MMSE-PIC detector for MI455X (gfx1250, wave32).
// One wave handles one batch element. gr = hr^T hr done with
// 8 chained V_WMMA_F32_16X16X4_F32 (fp32 WMMA, Gram-matrix trick:
// A and B fragments are identical per lane).

#define WAVES 4      // waves (batch elements) per block -> blockDim = 128
#define MDIM 16      // rx antennas
#define KST  8       // streams
#define NBITS 4      // bits/symbol (16-QAM)
#define CPTS 16      // constellation points
#define EPSV 1e-4f

typedef float v2f __attribute__((ext_vector_type(2)));
typedef float v8f __attribute__((ext_vector_type(8)));

struct WaveScratch {
  float Sr[256], Si[256];          // S, then complex Cholesky factor L
  float Wr[144], Wi[144];          // whitened [y' | H'] : 16 rows x 9 cols
  float HR[512];                   // real-valued channel [32][16]
  float GR[256];                   // gr = HR^T HR [16][16]
  float AUG[512];                  // Gauss-Jordan augmented [16][32]
  float ymfr[8], ymfi[8];          // matched filter h'^H y'
  float gxr[8], gxi[8];            // G * x_hat
  float logits[128];               // symbol logits [8][16]
  float xr[8], xi[8], varx[8];     // soft symbol moments
  float ypr_[64], ypi_[64];        // y_pic [i_mf_row][j_stream]
  float mu16[16], xh16[16], noeff[8];
  float llrA[32], llrD[32];
};

__device__ __forceinline__ float logsig(float x) {
  // log sigmoid(x) = min(x,0) - log1p(exp(-|x|))
  return fminf(x, 0.0f) - log1pf(__expf(-fabsf(x)));
}

// Gray-labeled PAM level for 2 bits (t = b0b1): 00->1, 01->3, 10->-1, 11->-3
__device__ __forceinline__ float pamv(int t) {
  float mag = (t & 1) ? 3.0f : 1.0f;
  return ((t >> 1) & 1) ? -mag : mag;
}

__global__ void __launch_bounds__(WAVES * 32) mmse_pic_det(
    const float* __restrict__ y_r, const float* __restrict__ y_i,
    const float* __restrict__ h_r, const float* __restrict__ h_i,
    const float* __restrict__ s_r, const float* __restrict__ s_i,
    const float* __restrict__ prior, float* __restrict__ out, int B)
{
  __shared__ WaveScratch smem[WAVES];
  const int tid  = threadIdx.x;
  const int wv   = tid >> 5;
  const int lane = tid & 31;
  const int b    = blockIdx.x * WAVES + wv;
  const bool act = (b < B);
  WaveScratch& W = smem[wv];

  const float INVS10 = 0.316227766016838f;  // 1/sqrt(10) unit-energy 16QAM

  // ---------------- load noise covariance S and prior LLRs ----------------
  if (act) {
    __builtin_prefetch(&h_r[(size_t)b * MDIM * KST], 0, 0);
    __builtin_prefetch(&h_i[(size_t)b * MDIM * KST], 0, 0);
    for (int t = lane; t < 256; t += 32) {
      W.Sr[t] = s_r[(size_t)b * 256 + t];
      W.Si[t] = s_i[(size_t)b * 256 + t];
    }
    W.llrD[lane] = prior[(size_t)b * 32 + lane];
    W.llrA[lane] = 0.0f;
  }
  __syncthreads();

  // ---------------- complex Cholesky S = L L^H (lower, in LDS) ------------
  for (int k = 0; k < MDIM; k++) {
    if (lane == 0) {
      float d = sqrtf(fmaxf(W.Sr[k*16+k], 1e-20f));
      W.Sr[k*16+k] = d; W.Si[k*16+k] = 0.0f;
    }
    __syncthreads();
    if (lane > k && lane < 16) {
      float inv = 1.0f / W.Sr[k*16+k];
      W.Sr[lane*16+k] *= inv; W.Si[lane*16+k] *= inv;
    }
    __syncthreads();
    for (int t = lane; t < 256; t += 32) {
      int ii = t >> 4, jj = t & 15;
      if (ii > k && jj > k && jj <= ii) {
        float ar = W.Sr[ii*16+k], ai = W.Si[ii*16+k];
        float br = W.Sr[jj*16+k], bi = W.Si[jj*16+k];
        W.Sr[ii*16+jj] -= ar*br + ai*bi;   // -= L[ii][k]*conj(L[jj][k])
        W.Si[ii*16+jj] -= ai*br - ar*bi;
      }
    }
    __syncthreads();
  }

  // ------- forward substitution: solve L * [y'|H'] = [y|H] (9 columns) ----
  if (lane < 9) {
    for (int i = 0; i < MDIM; i++) {
      float rr = 0.0f, ri = 0.0f;
      if (act) {
        if (lane == 0) { rr = y_r[(size_t)b*MDIM + i]; ri = y_i[(size_t)b*MDIM + i]; }
        else {
          rr = h_r[((size_t)b*MDIM + i)*KST + (lane-1)];
          ri = h_i[((size_t)b*MDIM + i)*KST + (lane-1)];
        }
      }
      for (int j = 0; j < i; j++) {
        float lr = W.Sr[i*16+j], li = W.Si[i*16+j];
        float wr = W.Wr[j*9+lane], wi = W.Wi[j*9+lane];
        rr -= lr*wr - li*wi;
        ri -= lr*wi + li*wr;
      }
      float d = 1.0f / W.Sr[i*16+i];
      W.Wr[i*9+lane] = rr * d; W.Wi[i*9+lane] = ri * d;
    }
  }
  __syncthreads();

  // ---------------- matched filter y_mf = H'^H y' -------------------------
  if (lane < 8) {
    float re = 0.0f, im = 0.0f;
    for (int m = 0; m < MDIM; m++) {
      float hr = W.Wr[m*9+1+lane], hi = W.Wi[m*9+1+lane];
      float yr = W.Wr[m*9],        yi = W.Wi[m*9];
      re += hr*yr + hi*yi;
      im += hr*yi - hi*yr;
    }
    W.ymfr[lane] = re; W.ymfi[lane] = im;
  }

  // ---------------- real-valued channel HR [32][16] -----------------------
  // HR = [[Re H', -Im H'], [Im H', Re H']]
  for (int t = 0; t < 16; t++) {
    int idx = t*32 + lane;
    int r = idx >> 4, c = idx & 15;
    int m = r & 15, k8 = c & 7;
    float hr = W.Wr[m*9+1+k8], hi = W.Wi[m*9+1+k8];
    float v;
    if (r < 16) v = (c < 8) ? hr : -hi;
    else        v = (c < 8) ? hi :  hr;
    W.HR[idx] = v;
  }
  __syncthreads();

  // ------- GR = HR^T * HR via 8 chained V_WMMA_F32_16X16X4_F32 ------------
  // D[i][j] += sum_{k<4} A[i][k]*B[k][j] with A[i][k]=HR[m0+k][i],
  // B[k][j]=HR[m0+k][j]. Per the ISA VGPR layouts the per-lane A and B
  // fragments are identical for a Gram matrix -> one v2f load feeds both.
  {
    v8f acc = {0.f,0.f,0.f,0.f,0.f,0.f,0.f,0.f};
    const int i16 = lane & 15;
    const int ks  = (lane >> 4) << 1;   // 0 for lanes 0-15, 2 for lanes 16-31
    #pragma unroll
    for (int m0 = 0; m0 < 32; m0 += 4) {
      v2f ab;
      ab.x = W.HR[(m0+ks  )*16 + i16];
      ab.y = W.HR[(m0+ks+1)*16 + i16];
      acc = __builtin_amdgcn_wmma_f32_16x16x4_f32(
          /*neg_a=*/false, ab, /*neg_b=*/false, ab,
          /*c_mod=*/(short)0, acc, /*reuse_a=*/false, /*reuse_b=*/false);
    }
    const int rbase = (lane >> 4) << 3;  // C/D layout: lanes16-31 hold M+8
    #pragma unroll
    for (int r = 0; r < 8; r++)
      W.GR[(r + rbase)*16 + i16] = acc[r];
  }
  __syncthreads();

  // ---------------- 2 MMSE-PIC iterations ---------------------------------
  for (int it = 0; it < 2; it++) {
    W.llrA[lane] = W.llrD[lane];             // llr_a = llr_d
    __syncthreads();

    // symbol logits[k][c] = sum_b log_sigmoid(sign_cb * llr_a[k][b])
    for (int t = 0; t < 4; t++) {
      int idx = t*32 + lane;
      int k = idx >> 4, c = idx & 15;
      float s = 0.0f;
      #pragma unroll
      for (int bb = 0; bb < NBITS; bb++) {
        float sg = ((c >> (3-bb)) & 1) ? 1.0f : -1.0f;
        s += logsig(sg * W.llrA[k*NBITS+bb]);
      }
      W.logits[idx] = s;
    }
    __syncthreads();

    // soft-symbol moments (softmax over constellation) per stream
    if (lane < 8) {
      int k = lane;
      float mx = -3.0e38f;
      for (int c = 0; c < CPTS; c++) mx = fmaxf(mx, W.logits[k*16+c]);
      float ssum = 0.f, sxr = 0.f, sxi = 0.f, sp2 = 0.f;
      for (int c = 0; c < CPTS; c++) {
        float p  = __expf(W.logits[k*16+c] - mx);
        float pr = pamv((c>>2)&3)*INVS10, pi = pamv(c&3)*INVS10;
        ssum += p; sxr += p*pr; sxi += p*pi; sp2 += p*(pr*pr+pi*pi);
      }
      float inv = 1.0f/ssum;
      float xr = sxr*inv, xi = sxi*inv;
      W.xr[k] = xr; W.xi[k] = xi;
      W.varx[k] = sp2*inv - (xr*xr + xi*xi);
    }
    __syncthreads();

    // gx = G * x_hat  (Re G = GR[0:8,0:8], Im G = GR[8:16,0:8])
    if (lane < 8) {
      int i = lane;
      float gr_ = 0.f, gi_ = 0.f;
      for (int l = 0; l < 8; l++) {
        float gre = W.GR[i*16+l], gim = W.GR[(i+8)*16+l];
        gr_ += gre*W.xr[l] - gim*W.xi[l];
        gi_ += gre*W.xi[l] + gim*W.xr[l];
      }
      W.gxr[i] = gr_; W.gxi[i] = gi_;
    }
    __syncthreads();

    // parallel interference cancellation y_pic[i][j]
    for (int t = 0; t < 2; t++) {
      int e = t*32 + lane;
      int i = e >> 3, j = e & 7;
      float gre = W.GR[i*16+j], gim = W.GR[(i+8)*16+j];
      W.ypr_[e] = W.ymfr[i] + (gre*W.xr[j] - gim*W.xi[j]) - W.gxr[i];
      W.ypi_[e] = W.ymfi[i] + (gre*W.xi[j] + gim*W.xr[j]) - W.gxi[i];
    }

    // AUG = [ GR*diag(v2) + I | I ],  v2 = [var, var]
    for (int t = 0; t < 16; t++) {
      int idx = t*32 + lane;
      int r = idx >> 5, c = idx & 31;
      float v;
      if (c < 16) {
        v = W.GR[r*16+c] * W.varx[c & 7];
        if (c == r) v += 1.0f;
      } else {
        v = (c - 16 == r) ? 1.0f : 0.0f;
      }
      W.AUG[idx] = v;
    }
    __syncthreads();

    // Gauss-Jordan inverse (A = I + G V is SPD-ish; no pivoting needed)
    for (int p = 0; p < 16; p++) {
      float inv = 1.0f / W.AUG[p*32+p];
      __syncthreads();
      W.AUG[p*32+lane] *= inv;            // normalize pivot row (32 cols)
      __syncthreads();
      float rp = W.AUG[p*32+lane];
      for (int i = 0; i < 16; i++) {
        if (i == p) continue;
        float f = W.AUG[i*32+p];          // load precedes store: wave-in-order
        W.AUG[i*32+lane] -= f * rp;
      }
      __syncthreads();
    }

    // mu = diag(A^-1 G); unbiased LMMSE estimate; post-eq noise
    if (lane < 16) {
      int i = lane, s8 = lane & 7;
      float mu = 0.f, sv = 0.f;
      for (int j = 0; j < 16; j++) {
        float ai_ = W.AUG[i*32+16+j];
        mu += ai_ * W.GR[i*16+j];
        float yv = (j < 8) ? W.ypr_[j*8 + s8] : W.ypi_[(j-8)*8 + s8];
        sv += ai_ * yv;
      }
      W.mu16[i] = mu;
      W.xh16[i] = sv / mu;
      if (i < 8) {
        float den = fmaxf(1.0f - W.varx[i]*mu, EPSV);
        W.noeff[i] = den / mu;            // no_eff = 1/var_p
      }
    }
    __syncthreads();

    // max-log demap with prior: lane = k*4 + bit
    {
      int k = lane >> 2, bb = lane & 3;
      float xr = W.xh16[k], xi = W.xh16[k+8];
      float ne = W.noeff[k];
      float m1 = -3.0e38f, m0 = -3.0e38f;
      for (int c = 0; c < CPTS; c++) {
        float pr = pamv((c>>2)&3)*INVS10, pi = pamv(c&3)*INVS10;
        float dr = xr - pr, di = xi - pi;
        float e = -(dr*dr + di*di)/ne + W.logits[k*16+c];
        if ((c >> (3-bb)) & 1) m1 = fmaxf(m1, e);
        else                   m0 = fmaxf(m0, e);
      }
      W.llrD[lane] = m1 - m0;
    }
    __syncthreads();
  }

  // extrinsic LLRs
  if (act) out[(size_t)b*32 + lane] = W.llrD[lane] - W.llrA[lane];
}

extern "C" void kernel_launch(void* const* d_in, const int* in_sizes, int n_in,
                              void* d_out, int out_size, void* d_ws, size_t ws_size,
                              hipStream_t stream) {
  const float* y_r   = (const float*)d_in[0];
  const float* y_i   = (const float*)d_in[1];
  const float* h_r   = (const float*)d_in[2];
  const float* h_i   = (const float*)d_in[3];
  const float* s_r   = (const float*)d_in[4];
  const float* s_i   = (const float*)d_in[5];
  const float* prior = (const float*)d_in[6];
  float* out = (float*)d_out;
  int B = in_sizes[0] / MDIM;                // in_sizes[0] = B*M
  int grid = (B + WAVES - 1) / WAVES;
  mmse_pic_det<<<grid, WAVES * 32, 0, stream>>>(y_r, y_i, h_r, h_i,
                                                s_r, s_i, prior, out, B);
}